// DispersionPotential_79860621902323
// MI455X (gfx1250) — compile-verified
//
#include <hip/hip_runtime.h>
#include <math.h>

#define BATCH 16
#define NATOM 256
#define MAXZ 87
#define NREF 7
#define K1C 16.0f
#define K3C 4.0f
#define A1C 0.566f
#define S8C 0.3908f
#define A2C 3.128f
#define BOHR_PER_NM 18.897259886f
#define HARTREE_TO_KJMOL 2625.4996f

typedef __attribute__((ext_vector_type(2))) float v2f;
typedef __attribute__((ext_vector_type(8))) float v8f;

// ---------------------------------------------------------------------------
// Pass 1: coordination numbers -> per-atom weights w[i,0..7] (7 used) and r4r2
// One block per batch system, 256 threads (8 waves).
// ---------------------------------------------------------------------------
__global__ __launch_bounds__(NATOM) void cn_kernel(
    const float* __restrict__ pos,      // (B*N,3)
    const int*   __restrict__ Zg,       // (B*N)
    const float* __restrict__ rcov,     // (MAXZ)
    const float* __restrict__ cn_ref,   // (MAXZ,NREF)
    const float* __restrict__ r4r2,     // (MAXZ)
    float* __restrict__ w_out,          // (B*N,8)
    float* __restrict__ r4_out)         // (B*N)
{
    const int b = blockIdx.x;
    const int i = threadIdx.x;

    __shared__ float sx[NATOM], sy[NATOM], sz[NATOM], src[NATOM];
    __shared__ int   sZ[NATOM];

    const size_t base = (size_t)b * NATOM;
    const int   Zi = Zg[base + i];
    const float x  = pos[(base + i) * 3 + 0];
    const float y  = pos[(base + i) * 3 + 1];
    const float z  = pos[(base + i) * 3 + 2];

    sx[i] = x; sy[i] = y; sz[i] = z;
    sZ[i] = Zi;
    src[i] = rcov[Zi];
    __syncthreads();

    const float rci  = src[i];
    const float cut2 = BOHR_PER_NM * BOHR_PER_NM;

    float cn = 0.0f;
    for (int j = 0; j < NATOM; ++j) {
        if (j == i) continue;
        const int Zj = sZ[j];
        const float dx = x - sx[j];
        const float dy = y - sy[j];
        const float dz = z - sz[j];
        const float d2 = dx * dx + dy * dy + dz * dz;
        if ((Zi > 0) && (Zj > 0) && (d2 <= cut2)) {
            const float d  = sqrtf(d2);
            const float cf = 1.0f / (1.0f + expf(-K1C * ((rci + src[j]) / d - 1.0f)));
            cn += cf;
        }
    }

    // Gaussian reference weights
    float gw[NREF];
    float s = 0.0f;
#pragma unroll
    for (int a = 0; a < NREF; ++a) {
        const float t = cn - cn_ref[Zi * NREF + a];
        gw[a] = expf(-K3C * t * t);
        s += gw[a];
    }
    const float inv = 1.0f / (s + 1e-20f);
#pragma unroll
    for (int a = 0; a < NREF; ++a)
        w_out[(base + i) * 8 + a] = gw[a] * inv;
    w_out[(base + i) * 8 + 7] = 0.0f;

    r4_out[base + i] = r4r2[Zi];
}

// ---------------------------------------------------------------------------
// Pass 2: per 16x16 tile, d^2 via V_WMMA_F32_16X16X4_F32, then pair energies.
// One wave (32 threads) per block. Grid: (256 tiles, B batches).
// ---------------------------------------------------------------------------
__global__ __launch_bounds__(32) void energy_kernel(
    const float* __restrict__ pos,      // (B*N,3)
    const int*   __restrict__ Zg,       // (B*N)
    const float* __restrict__ w_in,     // (B*N,8)
    const float* __restrict__ r4_in,    // (B*N)
    const float* __restrict__ c6_ref,   // (MAXZ,MAXZ,NREF,NREF)
    float* __restrict__ partial)        // (B, 256)
{
    const int tile = blockIdx.x;   // 0..255
    const int b    = blockIdx.y;   // batch
    const int ti   = tile >> 4;
    const int tj   = tile & 15;
    const int lane = threadIdx.x;  // 0..31 (wave32)
    const int i0   = ti * 16;
    const int j0   = tj * 16;

    __shared__ float pix[16], piy[16], piz[16], pi2[16];
    __shared__ float pjx[16], pjy[16], pjz[16], pj2[16];
    __shared__ int   szi[16], szj[16];
    __shared__ float red[32];

    const size_t base = (size_t)b * NATOM;
    {
        // lanes 0..15 stage the i-tile, lanes 16..31 the j-tile
        const int l = lane & 15;
        if (lane < 16) {
            const size_t gi = base + i0 + l;
            const float x = pos[gi * 3 + 0], y = pos[gi * 3 + 1], z = pos[gi * 3 + 2];
            pix[l] = x; piy[l] = y; piz[l] = z;
            pi2[l] = x * x + y * y + z * z;
            szi[l] = Zg[gi];
        } else {
            const size_t gj = base + j0 + l;
            const float x = pos[gj * 3 + 0], y = pos[gj * 3 + 1], z = pos[gj * 3 + 2];
            pjx[l] = x; pjy[l] = y; pjz[l] = z;
            pj2[l] = x * x + y * y + z * z;
            szj[l] = Zg[gj];
        }
    }
    __syncthreads();

    const int half = lane >> 4;    // 0: K=0/1 slice, 1: K=2/3 slice
    const int l15  = lane & 15;

    // A (16x4, f32): lanes 0-15 hold {K=0,K=1} for M=lane; lanes 16-31 hold
    // {K=2,K=3} for M=lane-16.  A = -2 * pos_i, K=3 column zero.
    v2f Amat;
    Amat.x = -2.0f * (half ? piz[l15] : pix[l15]);
    Amat.y = half ? 0.0f : (-2.0f * piy[l15]);

    // B (4x16, f32): lanes 0-15 hold {K=0,K=1} for N=lane; lanes 16-31 hold
    // {K=2,K=3} for N=lane-16.  B = pos_j^T, K=3 row zero.
    v2f Bmat;
    Bmat.x = half ? pjz[l15] : pjx[l15];
    Bmat.y = half ? 0.0f : pjy[l15];

    // C (16x16, f32): lane<16 -> M=v, N=lane; lane>=16 -> M=v+8, N=lane-16.
    v8f Cmat;
#pragma unroll
    for (int v = 0; v < 8; ++v)
        Cmat[v] = pi2[v + 8 * half] + pj2[l15];

    // D = A*B + C  ==  |p_i|^2 + |p_j|^2 - 2 p_i . p_j  ==  d^2 tile
    v8f d2v = __builtin_amdgcn_wmma_f32_16x16x4_f32(
        /*neg_a=*/false, Amat, /*neg_b=*/false, Bmat,
        /*c_mod=*/(short)0, Cmat, /*reuse_a=*/false, /*reuse_b=*/false);

    const float cut2 = BOHR_PER_NM * BOHR_PER_NM;
    float acc = 0.0f;

#pragma unroll
    for (int v = 0; v < 8; ++v) {
        const int m  = v + 8 * half;
        const int i  = i0 + m;
        const int j  = j0 + l15;
        const int Zi = szi[m];
        const int Zj = szj[l15];
        const float d2 = d2v[v];
        const bool ok = (i != j) && (Zi > 0) && (Zj > 0) && (d2 <= cut2);
        if (ok) {
            const float* wi = w_in + (base + i) * 8;
            const float* wj = w_in + (base + j) * 8;
            const float* cb = c6_ref + ((size_t)Zi * MAXZ + Zj) * (NREF * NREF);
            __builtin_prefetch(cb, 0, 3);   // global_prefetch_b8

            // c6ij = w_i^T * C6ref(Zi,Zj) * w_j   (49 MACs)
            float c6 = 0.0f;
#pragma unroll
            for (int a = 0; a < NREF; ++a) {
                float s = 0.0f;
#pragma unroll
                for (int q = 0; q < NREF; ++q)
                    s = fmaf(cb[a * NREF + q], wj[q], s);
                c6 = fmaf(wi[a], s, c6);
            }

            const float qq = 3.0f * r4_in[base + i] * r4_in[base + j];
            const float c8 = c6 * qq;
            const float r0 = A1C * sqrtf(qq) + A2C;
            const float d6 = d2 * d2 * d2;      // d^6 == (d^2)^3 exactly
            const float d8 = d6 * d2;
            const float r02 = r0 * r0;
            const float r06 = r02 * r02 * r02;
            const float r08 = r06 * r02;
            acc += c6 / (d6 + r06) + S8C * c8 / (d8 + r08);
        }
    }

    // Deterministic in-wave reduction via LDS (no float atomics)
    red[lane] = acc;
    __syncthreads();
    if (lane == 0) {
        float s = 0.0f;
#pragma unroll
        for (int k = 0; k < 32; ++k) s += red[k];
        partial[b * 256 + tile] = s;
    }
}

// ---------------------------------------------------------------------------
// Pass 3: reduce 256 tile partials per batch, scale, write output (B,1).
// ---------------------------------------------------------------------------
__global__ __launch_bounds__(256) void reduce_kernel(
    const float* __restrict__ partial, float* __restrict__ out)
{
    const int b = blockIdx.x;
    const int t = threadIdx.x;
    __shared__ float s[256];
    s[t] = partial[b * 256 + t];
    __syncthreads();
    for (int off = 128; off > 0; off >>= 1) {
        if (t < off) s[t] += s[t + off];
        __syncthreads();
    }
    if (t == 0)
        out[b] = s[0] * (-0.5f * HARTREE_TO_KJMOL);
}

// ---------------------------------------------------------------------------
extern "C" void kernel_launch(void* const* d_in, const int* in_sizes, int n_in,
                              void* d_out, int out_size, void* d_ws, size_t ws_size,
                              hipStream_t stream) {
    const float* pos    = (const float*)d_in[0];   // (B*N,3) f32
    const int*   Zg     = (const int*)d_in[1];     // (B*N)   i32
    // d_in[2] = atomic_subsystem_indices (unused; batches are contiguous)
    const float* rcov   = (const float*)d_in[3];   // (MAXZ)
    const float* r4r2   = (const float*)d_in[4];   // (MAXZ)
    const float* cn_ref = (const float*)d_in[5];   // (MAXZ,NREF)
    const float* c6ref  = (const float*)d_in[6];   // (MAXZ,MAXZ,NREF,NREF)
    float* out = (float*)d_out;

    float* w_ws  = (float*)d_ws;                     // B*N*8 floats
    float* r4_ws = w_ws + (size_t)BATCH * NATOM * 8; // B*N   floats
    float* part  = r4_ws + (size_t)BATCH * NATOM;    // B*256 floats

    cn_kernel<<<dim3(BATCH), dim3(NATOM), 0, stream>>>(
        pos, Zg, rcov, cn_ref, r4r2, w_ws, r4_ws);

    energy_kernel<<<dim3(256, BATCH), dim3(32), 0, stream>>>(
        pos, Zg, w_ws, r4_ws, c6ref, part);

    reduce_kernel<<<dim3(BATCH), dim3(256), 0, stream>>>(part, out);
}